// TempHypE_GNN_57397942944298
// MI455X (gfx1250) — compile-verified
//
#include <hip/hip_runtime.h>
#include <hip/hip_bf16.h>

typedef __attribute__((ext_vector_type(2))) float v2f;
typedef __attribute__((ext_vector_type(8))) float v8f;

// ---------------------------------------------------------------------------
// Degree / normalization
// ---------------------------------------------------------------------------
__global__ void deg_init_kernel(float* __restrict__ deg, int n) {
    int i = blockIdx.x * blockDim.x + threadIdx.x;
    if (i < n) deg[i] = 1.0f;           // self-loop contribution
}

__global__ void deg_count_kernel(const int* __restrict__ dst, float* __restrict__ deg, int e) {
    int i = blockIdx.x * blockDim.x + threadIdx.x;
    if (i < e) atomicAdd(&deg[dst[i]], 1.0f);
}

__global__ void deg_rsqrt_kernel(float* __restrict__ deg, int n) {
    int i = blockIdx.x * blockDim.x + threadIdx.x;
    if (i < n) deg[i] = rsqrtf(deg[i]); // deg >= 1 always (self loop)
}

// ---------------------------------------------------------------------------
// fp32 WMMA GEMM:  C[M x N] = (gather ? A[gather[m]] : A[m]) [M x K] @ B[K x N]
// One wave computes a 64x16 strip of C (4 stacked 16x16 WMMA tiles reusing B).
// M must be a multiple of 16 (50000 = 16 * 3125).
// ---------------------------------------------------------------------------
__global__ __launch_bounds__(256)
void gemm_wmma_f32(const float* __restrict__ A, const int* __restrict__ gather,
                   const float* __restrict__ B, float* __restrict__ C,
                   int M, int N, int K)
{
    const int wave  = (int)((blockIdx.x * blockDim.x + threadIdx.x) >> 5);
    const int lane  = threadIdx.x & 31;
    const int tilesN = N >> 4;
    const int mGroup = wave / tilesN;          // 64-row group
    const int nTile  = wave - mGroup * tilesN;
    const int half   = lane >> 4;              // 0: lanes 0-15, 1: lanes 16-31
    const int l      = lane & 15;

    const int row0 = mGroup * 64;
    if (row0 >= M) return;                     // uniform per wave

    // Resolve A-row pointers for the 4 sub-tiles (uniform validity per tile).
    const float* arow[4];
    bool tvalid[4];
    #pragma unroll
    for (int t = 0; t < 4; ++t) {
        int m = row0 + t * 16 + l;
        tvalid[t] = (row0 + t * 16) < M;       // whole 16-row tile valid or not
        int r = (m < M) ? m : 0;
        int src = gather ? gather[r] : r;
        arow[t] = A + (size_t)src * K;
    }
    const int n = nTile * 16 + l;

    v8f acc[4] = {v8f{}, v8f{}, v8f{}, v8f{}};

    for (int k0 = 0; k0 < K; k0 += 4) {
        const int kb = k0 + 2 * half;          // per ISA A/B fragment layout
        v2f bfrag;
        bfrag.x = B[(size_t)(kb + 0) * N + n];
        bfrag.y = B[(size_t)(kb + 1) * N + n];
        #pragma unroll
        for (int t = 0; t < 4; ++t) {
            v2f afrag;
            afrag.x = arow[t][kb + 0];
            afrag.y = arow[t][kb + 1];
            acc[t] = __builtin_amdgcn_wmma_f32_16x16x4_f32(
                false, afrag, false, bfrag, (short)0, acc[t], false, false);
        }
    }

    #pragma unroll
    for (int t = 0; t < 4; ++t) {
        if (!tvalid[t]) continue;              // uniform branch
        const int mbase = row0 + t * 16 + 8 * half;
        #pragma unroll
        for (int r = 0; r < 8; ++r) {
            C[(size_t)(mbase + r) * N + n] = acc[t][r];
        }
    }
}

// ---------------------------------------------------------------------------
// Aggregation: out[d] = sum_{e: dst=d} h[src] * dinv[src]*dinv[dst]
//            + h[d]*dinv[d]^2 (self loop) + bias
// ---------------------------------------------------------------------------
__global__ void agg_init_kernel(const float* __restrict__ h, const float* __restrict__ dinv,
                                const float* __restrict__ bias, float* __restrict__ out,
                                int n, int F)
{
    int i = blockIdx.x * blockDim.x + threadIdx.x;
    int total = n * F;
    if (i >= total) return;
    int node = i / F;
    int f    = i - node * F;
    float w = dinv[node];
    out[i] = h[i] * w * w + bias[f];
}

__global__ void agg_edges_kernel(const float* __restrict__ h, const int* __restrict__ src,
                                 const int* __restrict__ dst, const float* __restrict__ dinv,
                                 float* __restrict__ out, int E, int F)
{
    const int tpe = F >> 2;                       // threads per edge (float4 each)
    long long gid = (long long)blockIdx.x * blockDim.x + threadIdx.x;
    long long e = gid / tpe;
    int f4 = (int)(gid - e * tpe);
    if (e >= E) return;
    int s = src[e], d = dst[e];
    float w = dinv[s] * dinv[d];
    const float4 hv = ((const float4*)(h + (size_t)s * F))[f4];
    float* o = out + (size_t)d * F + (size_t)f4 * 4;
    atomicAdd(o + 0, hv.x * w);
    atomicAdd(o + 1, hv.y * w);
    atomicAdd(o + 2, hv.z * w);
    atomicAdd(o + 3, hv.w * w);
}

// ---------------------------------------------------------------------------
// Fused (optional ReLU) + LayerNorm, one wave32 per node.
// ---------------------------------------------------------------------------
__global__ __launch_bounds__(256)
void relu_ln_kernel(const float* __restrict__ x, const float* __restrict__ gamma,
                    const float* __restrict__ beta, float* __restrict__ y,
                    int n, int F, int doRelu)
{
    const int wave = (int)((blockIdx.x * blockDim.x + threadIdx.x) >> 5);
    const int lane = threadIdx.x & 31;
    if (wave >= n) return;
    const float* xr = x + (size_t)wave * F;
    const int per = F >> 5;                       // 8 (F=256) or 4 (F=128)

    float v[8];
    float sum = 0.0f;
    for (int i = 0; i < per; ++i) {
        float t = xr[lane + i * 32];
        if (doRelu) t = fmaxf(t, 0.0f);
        v[i] = t;
        sum += t;
    }
    #pragma unroll
    for (int off = 16; off; off >>= 1) sum += __shfl_xor(sum, off, 32);
    const float mu = sum / (float)F;

    float vs = 0.0f;
    for (int i = 0; i < per; ++i) { float d = v[i] - mu; vs += d * d; }
    #pragma unroll
    for (int off = 16; off; off >>= 1) vs += __shfl_xor(vs, off, 32);
    const float rstd = rsqrtf(vs / (float)F + 1e-5f);

    float* yr = y + (size_t)wave * F;
    for (int i = 0; i < per; ++i) {
        int f = lane + i * 32;
        yr[f] = (v[i] - mu) * rstd * gamma[f] + beta[f];
    }
}

// ---------------------------------------------------------------------------
// Host-side orchestration
// ---------------------------------------------------------------------------
extern "C" void kernel_launch(void* const* d_in, const int* in_sizes, int n_in,
                              void* d_out, int out_size, void* d_ws, size_t ws_size,
                              hipStream_t stream)
{
    const int*   x_ids  = (const int*)  d_in[0];
    const int*   eidx   = (const int*)  d_in[1];
    const float* embed  = (const float*)d_in[2];
    const float* W1     = (const float*)d_in[3];
    const float* b1     = (const float*)d_in[4];
    const float* g1     = (const float*)d_in[5];
    const float* be1    = (const float*)d_in[6];
    const float* W2     = (const float*)d_in[7];
    const float* b2     = (const float*)d_in[8];
    const float* g2     = (const float*)d_in[9];
    const float* be2    = (const float*)d_in[10];

    const int N  = in_sizes[0];              // 50000
    const int E  = in_sizes[1] / 2;          // 800000
    const int F0 = in_sizes[2] / N;          // 128 (embed dim)
    const int H  = in_sizes[4];              // 256 (hidden)
    const int F2 = in_sizes[8];              // 128 (out dim)

    const int* src = eidx;
    const int* dst = eidx + E;

    // Workspace layout (floats): dinv | bufA (N*H) | bufB (N*H)
    float* dinv = (float*)d_ws;
    size_t off  = ((size_t)N + 255) & ~(size_t)255;
    float* bufA = dinv + off;
    float* bufB = bufA + (size_t)N * H;

    const int TB = 256;

    // --- degree normalization ---
    deg_init_kernel <<<(N + TB - 1) / TB, TB, 0, stream>>>(dinv, N);
    deg_count_kernel<<<(E + TB - 1) / TB, TB, 0, stream>>>(dst, dinv, E);
    deg_rsqrt_kernel<<<(N + TB - 1) / TB, TB, 0, stream>>>(dinv, N);

    auto launch_gemm = [&](const float* A, const int* gidx, const float* B,
                           float* C, int M, int Nn, int Kk) {
        int tilesN = Nn >> 4;
        int mGroups = (M + 63) >> 6;
        long long waves = (long long)mGroups * tilesN;
        int blocks = (int)((waves * 32 + TB - 1) / TB);
        gemm_wmma_f32<<<blocks, TB, 0, stream>>>(A, gidx, B, C, M, Nn, Kk);
    };

    // --- layer 1: h1 = embed[x_ids] @ W1 ---
    launch_gemm(embed, x_ids, W1, bufA, N, H, F0);

    // agg1: bufB = scatter(h1) + self + b1
    {
        int total = N * H;
        agg_init_kernel<<<(total + TB - 1) / TB, TB, 0, stream>>>(bufA, dinv, b1, bufB, N, H);
        long long threads = (long long)E * (H >> 2);
        int blocks = (int)((threads + TB - 1) / TB);
        agg_edges_kernel<<<blocks, TB, 0, stream>>>(bufA, src, dst, dinv, bufB, E, H);
    }

    // relu + LN -> bufA
    relu_ln_kernel<<<((long long)N * 32 + TB - 1) / TB, TB, 0, stream>>>(bufB, g1, be1, bufA, N, H, 1);

    // --- layer 2: h2 = bufA @ W2 -> bufB (N x F2) ---
    launch_gemm(bufA, nullptr, W2, bufB, N, F2, H);

    // agg2: bufA = scatter(h2) + self + b2
    {
        int total = N * F2;
        agg_init_kernel<<<(total + TB - 1) / TB, TB, 0, stream>>>(bufB, dinv, b2, bufA, N, F2);
        long long threads = (long long)E * (F2 >> 2);
        int blocks = (int)((threads + TB - 1) / TB);
        agg_edges_kernel<<<blocks, TB, 0, stream>>>(bufB, src, dst, dinv, bufA, E, F2);
    }

    // final LN -> d_out
    relu_ln_kernel<<<((long long)N * 32 + TB - 1) / TB, TB, 0, stream>>>(bufA, g2, be2, (float*)d_out, N, F2, 0);
}